// BiLstmCRF_29815662969207
// MI455X (gfx1250) — compile-verified
//
#include <hip/hip_runtime.h>
#include <stdint.h>

// ---------------------------------------------------------------------------
// BiLSTM-CRF forward log-likelihood for MI455X (gfx1250, wave32, WMMA + TDM).
//
//   1) lstm_scan_kernel  (persistent, grid=16x2, device-wide barrier per step)
//        gates = [emb-gathered x_t ; h_{t-1}] * [Wih^T ; Whh^T] + b
//        via V_WMMA_F32_16X16X4_F32 (exact f32 matrix path on CDNA5).
//        Wih/Whh slices staged into LDS once via TENSOR_LOAD_TO_LDS (TDM),
//        using the TDM pad feature to skew LDS rows (bank-conflict-free).
//        Inner k-loop is branchless: out-of-length lanes read a zeroed pad
//        row through a per-step pointer select instead of exec-mask dancing.
//        c fragments live in registers across all 512 steps; h state is
//        double-buffered in global (L2-resident).
//   2) em_kernel    : em = [h_f ; rev(h_b)] @ W_out^T + b_out
//   3) score_kernel : CRF gold-path score per sequence
//   4) crf_alpha    : CRF forward algorithm (logsumexp over K=9)
//   5) final_kernel : sum(score - logZ) -> scalar
// ---------------------------------------------------------------------------

typedef __attribute__((ext_vector_type(2))) float v2f;
typedef __attribute__((ext_vector_type(8))) float v8f;
typedef __attribute__((ext_vector_type(4))) unsigned int v4u;
typedef __attribute__((ext_vector_type(4))) int v4i;
typedef __attribute__((ext_vector_type(8))) int v8i;

constexpr int E = 256, H = 256, K = 9, B = 128, T = 512;
constexpr int NWG = 16;                // workgroups per direction (16 h-cols each)
constexpr int TOTAL_WG = NWG * 2;
constexpr int LDSP = 258;              // padded LDS row stride (floats): 2-DWORD skew

static __device__ __forceinline__ float sigmoidf_(float x) {
  return 1.0f / (1.0f + __expf(-x));
}
static __device__ __forceinline__ float tanhf_(float x) {
  return 1.0f - 2.0f / (__expf(2.0f * x) + 1.0f);
}

// D = A(16x4 f32) * B(4x16 f32) + C(16x16 f32), full f32 precision.
static __device__ __forceinline__ v8f wmma_f32(v2f a, v2f b, v8f c) {
  return __builtin_amdgcn_wmma_f32_16x16x4_f32(
      /*neg_a=*/false, a, /*neg_b=*/false, b,
      /*c_mod=*/(short)0, c, /*reuse_a=*/false, /*reuse_b=*/false);
}

// Toolchain arity split for the TDM builtin (ROCm7.2 clang-22: 5 args,
// amdgpu-toolchain clang-23: 6 args).
#if defined(__clang_major__) && (__clang_major__ >= 23)
#define TDM_LOAD(g0, g1, g2, g3) \
  __builtin_amdgcn_tensor_load_to_lds((g0), (g1), (g2), (g3), \
                                      (v8i){0, 0, 0, 0, 0, 0, 0, 0}, 0)
#else
#define TDM_LOAD(g0, g1, g2, g3) \
  __builtin_amdgcn_tensor_load_to_lds((g0), (g1), (g2), (g3), 0)
#endif

// TDM: DMA a 16-row x 256-element f32 tile (row stride 256 elements in global)
// into LDS, inserting 2 pad DWORDs after every 256 DWORDs so LDS rows land at a
// 258-float stride (bank-conflict-free B-fragment reads).
// D# packing per CDNA5 ISA.
static __device__ __forceinline__ void tdm_load_tile_256x16(const float* gsrc,
                                                            const float* lds_dst) {
  const uint64_t ga = (uint64_t)(uintptr_t)gsrc;
  const uint32_t la = (uint32_t)(uintptr_t)lds_dst;  // generic->LDS offset (addr[31:0])
  // group0: count=1 | lds_addr | global_addr[56:0] | type=2
  v4u g0 = {1u, la, (uint32_t)ga,
            (uint32_t)((ga >> 32) & 0x01FFFFFFu) | 0x80000000u};
  // group1: data_size=4B, pad_enable, pad_interval=256 DW, pad_amount=2 DW,
  //         tensor_dim0=256, tensor_dim1=16, tile_dim0=256, tile_dim1=16,
  //         tensor_dim0_stride=256, tensor_dim1_stride=0 (2D tile).
  v8i g1 = {(int)((2u << 16) | (1u << 20) | (7u << 22) | (1u << 25)),
            (int)(256u << 16),   // tensor_dim0[15:0] at bits 63:48
            (int)(16u << 16),    // tensor_dim0[31:16]=0 | tensor_dim1[15:0]
            (int)(256u << 16),   // tensor_dim1[31:16]=0 | tile_dim0
            (int)16u,            // tile_dim1 | tile_dim2=0
            (int)256u,           // tensor_dim0_stride[31:0]
            0, 0};
  v4i z4 = {0, 0, 0, 0};
  TDM_LOAD(g0, g1, z4, z4);
}

// ---------------------------------------------------------------------------
// Kernel 1: persistent bidirectional LSTM scan.
//   blockIdx.x = h-column slice (16 cols), blockIdx.y = direction.
//   wave w owns batch M-tile w (rows 16w..16w+15) for all 4 gates of its slice.
// ---------------------------------------------------------------------------
__global__ __launch_bounds__(256)
void lstm_scan_kernel(const int* __restrict__ sent,
                      const int* __restrict__ lens,
                      const float* __restrict__ emb,
                      const float* __restrict__ Wih_f,
                      const float* __restrict__ Whh_f,
                      const float* __restrict__ b_f,
                      const float* __restrict__ Wih_b,
                      const float* __restrict__ Whh_b,
                      const float* __restrict__ b_b,
                      float* __restrict__ h_f_out,
                      float* __restrict__ h_b_out,
                      float* __restrict__ h_state,   // [2 dir][2 buf][B*H], zeroed
                      int* __restrict__ sync_cnt,    // zeroed
                      int* __restrict__ sync_gen,    // zeroed
                      const float* __restrict__ zpad)// 256 zeroed floats
{
  const int wg   = blockIdx.x;      // 0..15
  const int dir  = blockIdx.y;      // 0 = fwd, 1 = bwd
  const int tid  = threadIdx.x;
  const int wv   = tid >> 5;        // 0..7 : batch M-tile
  const int lane = tid & 31;
  const int n    = lane & 15;
  const int half = lane >> 4;       // K half-select for A/B fragments

  const float* Wih = dir ? Wih_b : Wih_f;
  const float* Whh = dir ? Whh_b : Whh_f;
  const float* bia = dir ? b_b   : b_f;
  float* h_out = dir ? h_b_out : h_f_out;   // bwd stored in scan (reversed) order
  float* hs    = h_state + (size_t)dir * 2 * B * H;

  const int colbase = wg * 16;
  const int col     = colbase + n;  // this lane's h-column

  // Time-invariant weight slices: [m][g][n][k] with 2-DWORD row skew.
  //   m=0: Wih slice, m=1: Whh slice.  ~129 KB of the 320 KB WGP LDS.
  __shared__ __align__(16) float sW[2][4][16][LDSP];

  // One wave issues the 8 tile DMAs (TDM is a wave-level op; EXEC ignored).
  if (wv == 0) {
#pragma unroll
    for (int m = 0; m < 2; ++m) {
      const float* src = m ? Whh : Wih;
#pragma unroll
      for (int g = 0; g < 4; ++g)
        tdm_load_tile_256x16(src + (size_t)(g * H + colbase) * E,
                             &sW[m][g][0][0]);
    }
    __builtin_amdgcn_s_wait_tensorcnt(0);
  }
  __syncthreads();

  // Per-lane constants.
  const int rowA = wv * 16 + n;     // batch row this lane supplies as A operand
  const int lenA = lens[rowA];
  float bias[4];
#pragma unroll
  for (int g = 0; g < 4; ++g) bias[g] = bia[g * H + col];
  int rowC[8], lenC[8];             // C-fragment row mapping: M = r + 8*half
#pragma unroll
  for (int r = 0; r < 8; ++r) {
    rowC[r] = wv * 16 + r + 8 * half;
    lenC[r] = lens[rowC[r]];
  }

  float cfrag[8];                   // cell state tile, register-resident for all T
#pragma unroll
  for (int r = 0; r < 8; ++r) cfrag[r] = 0.0f;

  for (int t = 0; t < T; ++t) {
    const int cur = t & 1, nxt = cur ^ 1;
    const float* hrow = hs + (size_t)cur * B * H + (size_t)rowA * H;

    // Embedding-gather A row for the input transform (direction aware).
    // Out-of-length lanes read the zeroed pad row: branchless k-loop below.
    bool xok; int teff;
    if (dir == 0) { xok = true;          teff = t; }
    else          { xok = (t < lenA);    teff = xok ? (lenA - 1 - t) : 0; }
    const int tok = sent[(size_t)rowA * T + teff];
    const float* xrow = xok ? (emb + (size_t)tok * E) : zpad;

    v8f acc0 = {0,0,0,0,0,0,0,0}, acc1 = acc0, acc2 = acc0, acc3 = acc0;

    // gates += x_t @ Wih^T   (K = 256; A from L2-hot emb, B from LDS)
#pragma unroll 4
    for (int kk = 0; kk < E; kk += 4) {
      const int kl = kk + 2 * half;
      v2f a  = *(const v2f*)(xrow + kl);
      v2f w0 = *(const v2f*)(&sW[0][0][n][kl]);
      v2f w1 = *(const v2f*)(&sW[0][1][n][kl]);
      v2f w2 = *(const v2f*)(&sW[0][2][n][kl]);
      v2f w3 = *(const v2f*)(&sW[0][3][n][kl]);
      acc0 = wmma_f32(a, w0, acc0);
      acc1 = wmma_f32(a, w1, acc1);
      acc2 = wmma_f32(a, w2, acc2);
      acc3 = wmma_f32(a, w3, acc3);
    }
    // gates += h_{t-1} @ Whh^T   (K = 256; A from L2-hot h state, B from LDS)
#pragma unroll 4
    for (int kk = 0; kk < H; kk += 4) {
      const int kl = kk + 2 * half;
      v2f a  = *(const v2f*)(hrow + kl);
      v2f w0 = *(const v2f*)(&sW[1][0][n][kl]);
      v2f w1 = *(const v2f*)(&sW[1][1][n][kl]);
      v2f w2 = *(const v2f*)(&sW[1][2][n][kl]);
      v2f w3 = *(const v2f*)(&sW[1][3][n][kl]);
      acc0 = wmma_f32(a, w0, acc0);
      acc1 = wmma_f32(a, w1, acc1);
      acc2 = wmma_f32(a, w2, acc2);
      acc3 = wmma_f32(a, w3, acc3);
    }

    // Pointwise LSTM cell on the 16x16 tile (i,f,g,o fragments align element-wise).
    const float* hcur = hs + (size_t)cur * B * H;
    float*       hnxt = hs + (size_t)nxt * B * H;
#pragma unroll
    for (int r = 0; r < 8; ++r) {
      float iv = sigmoidf_(acc0[r] + bias[0]);
      float fv = sigmoidf_(acc1[r] + bias[1]);
      float gv = tanhf_   (acc2[r] + bias[2]);
      float ov = sigmoidf_(acc3[r] + bias[3]);
      float cn = fv * cfrag[r] + iv * gv;
      float hn = ov * tanhf_(cn);
      const bool  valid = (t < lenC[r]);
      const float hold  = hcur[(size_t)rowC[r] * H + col];
      cfrag[r] = valid ? cn : cfrag[r];
      hnxt[(size_t)rowC[r] * H + col] = valid ? hn : hold;       // carry state
      h_out[((size_t)rowC[r] * T + t) * H + col] = valid ? hn : 0.0f;  // ys = h_new*v
    }

    // Device-wide barrier between timesteps (cumulative counter + generation).
    __syncthreads();
    if (tid == 0) {
      __threadfence();
      int my = atomicAdd(sync_cnt, 1);
      if (my == TOTAL_WG * (t + 1) - 1) {
        __threadfence();
        atomicExch(sync_gen, t + 1);
      } else {
        while (atomicAdd(sync_gen, 0) < t + 1) __builtin_amdgcn_s_sleep(2);
      }
    }
    __syncthreads();
    __threadfence();
  }
}

// ---------------------------------------------------------------------------
// Kernel 2: em[b,t,k] = W_out[k,:] . [h_f[b,t] ; rev(h_b)[b,t]] + b_out[k]
// One wave per (b,t); 16 concat elements per lane; wave shuffle reduction.
// ---------------------------------------------------------------------------
__global__ __launch_bounds__(256)
void em_kernel(const float* __restrict__ h_f, const float* __restrict__ h_b,
               const int* __restrict__ lens,
               const float* __restrict__ W_out, const float* __restrict__ b_out,
               float* __restrict__ em)
{
  const int wv = threadIdx.x >> 5, lane = threadIdx.x & 31;
  const int bt = blockIdx.x * 8 + wv;
  const int b = bt / T, t = bt % T;
  const int len = lens[b];
  const bool ok = t < len;
  const int  tr = ok ? (len - 1 - t) : 0;

  float hc[16];
#pragma unroll
  for (int i = 0; i < 8; ++i)
    hc[i] = h_f[((size_t)b * T + t) * H + lane + 32 * i];
#pragma unroll
  for (int i = 0; i < 8; ++i)
    hc[8 + i] = ok ? h_b[((size_t)b * T + tr) * H + lane + 32 * i] : 0.0f;

  for (int k = 0; k < K; ++k) {
    float s = 0.0f;
#pragma unroll
    for (int i = 0; i < 16; ++i)
      s += W_out[(size_t)k * (2 * H) + lane + 32 * i] * hc[i];
#pragma unroll
    for (int off = 16; off; off >>= 1) s += __shfl_xor(s, off, 32);
    if (lane == 0) em[(size_t)bt * K + k] = s + b_out[k];
  }
}

// ---------------------------------------------------------------------------
// Kernel 3: CRF gold-path score per sequence (one wave per b).
// ---------------------------------------------------------------------------
__global__ __launch_bounds__(256)
void score_kernel(const int* __restrict__ tags, const int* __restrict__ lens,
                  const float* __restrict__ em,
                  const float* __restrict__ start, const float* __restrict__ endv,
                  const float* __restrict__ trans, float* __restrict__ score)
{
  const int wv = threadIdx.x >> 5, lane = threadIdx.x & 31;
  const int b = blockIdx.x * 8 + wv;
  const int len = lens[b];
  float s = 0.0f;
  for (int t = 1 + lane; t < len; t += 32) {
    int tg = tags[(size_t)b * T + t];
    int tp = tags[(size_t)b * T + t - 1];
    s += trans[tp * K + tg] + em[((size_t)b * T + t) * K + tg];
  }
#pragma unroll
  for (int off = 16; off; off >>= 1) s += __shfl_xor(s, off, 32);
  if (lane == 0) {
    int t0 = tags[(size_t)b * T];
    int tl = tags[(size_t)b * T + len - 1];
    score[b] = s + start[t0] + em[((size_t)b * T) * K + t0] + endv[tl];
  }
}

// ---------------------------------------------------------------------------
// Kernel 4: CRF forward algorithm (logZ). 32 sequences per block, thread=(b,k).
// ---------------------------------------------------------------------------
__global__ __launch_bounds__(288)
void crf_alpha_kernel(const float* __restrict__ em, const int* __restrict__ lens,
                      const float* __restrict__ start, const float* __restrict__ endv,
                      const float* __restrict__ trans, float* __restrict__ logZ)
{
  __shared__ float sa[32][K];
  __shared__ float st[K][K];
  __shared__ float sv[32][K];
  const int tid = threadIdx.x;
  const int bl = tid / K, k = tid % K;
  const int b = blockIdx.x * 32 + bl;
  if (tid < K * K) st[tid / K][tid % K] = trans[tid];
  const int len = lens[b];
  sa[bl][k] = start[k] + em[((size_t)b * T) * K + k];
  __syncthreads();

  for (int t = 1; t < T; ++t) {
    float m = -1e30f;
#pragma unroll
    for (int j = 0; j < K; ++j) m = fmaxf(m, sa[bl][j] + st[j][k]);
    float s = 0.0f;
#pragma unroll
    for (int j = 0; j < K; ++j) s += __expf(sa[bl][j] + st[j][k] - m);
    float nxt = m + __logf(s) + em[((size_t)b * T + t) * K + k];
    __syncthreads();
    if (t < len) sa[bl][k] = nxt;
    __syncthreads();
  }

  sv[bl][k] = sa[bl][k] + endv[k];
  __syncthreads();
  if (k == 0) {
    float m = -1e30f;
    for (int j = 0; j < K; ++j) m = fmaxf(m, sv[bl][j]);
    float s = 0.0f;
    for (int j = 0; j < K; ++j) s += __expf(sv[bl][j] - m);
    logZ[b] = m + __logf(s);
  }
}

// ---------------------------------------------------------------------------
// Kernel 5: out = sum_b(score[b] - logZ[b])
// ---------------------------------------------------------------------------
__global__ __launch_bounds__(128)
void final_kernel(const float* __restrict__ score, const float* __restrict__ logZ,
                  float* __restrict__ out)
{
  __shared__ float red[128];
  const int tid = threadIdx.x;
  red[tid] = score[tid] - logZ[tid];
  __syncthreads();
  for (int s = 64; s; s >>= 1) {
    if (tid < s) red[tid] += red[tid + s];
    __syncthreads();
  }
  if (tid == 0) out[0] = red[0];
}

// ---------------------------------------------------------------------------
extern "C" void kernel_launch(void* const* d_in, const int* in_sizes, int n_in,
                              void* d_out, int out_size, void* d_ws, size_t ws_size,
                              hipStream_t stream)
{
  (void)in_sizes; (void)n_in; (void)out_size; (void)ws_size;

  const int*   sent  = (const int*)d_in[0];
  const int*   tags  = (const int*)d_in[1];
  const int*   lens  = (const int*)d_in[2];
  const float* emb   = (const float*)d_in[3];
  const float* Wih_f = (const float*)d_in[4];
  const float* Whh_f = (const float*)d_in[5];
  const float* b_f   = (const float*)d_in[6];
  const float* Wih_b = (const float*)d_in[7];
  const float* Whh_b = (const float*)d_in[8];
  const float* b_b   = (const float*)d_in[9];
  const float* W_out = (const float*)d_in[10];
  const float* b_out = (const float*)d_in[11];
  const float* start = (const float*)d_in[12];
  const float* endv  = (const float*)d_in[13];
  const float* trans = (const float*)d_in[14];

  // Workspace layout (floats): ~137 MB total.
  float* ws    = (float*)d_ws;
  float* h_f   = ws;                              // B*T*H
  float* h_b   = h_f  + (size_t)B * T * H;        // B*T*H (scan order)
  float* h_st  = h_b  + (size_t)B * T * H;        // 2 dir * 2 buf * B*H
  float* em    = h_st + (size_t)4 * B * H;        // B*T*K
  float* score = em   + (size_t)B * T * K;        // B
  float* logZ  = score + B;                       // B
  int*   syncp = (int*)(logZ + B);                // cnt, gen, pad, pad
  float* zpad  = (float*)(syncp + 4);             // 256 zeroed floats

  // Re-arm persistent-kernel state every call (workspace is not re-zeroed by
  // the harness between timed replays). Sync words + zero pad are contiguous.
  hipMemsetAsync(h_st, 0, (size_t)4 * B * H * sizeof(float), stream);
  hipMemsetAsync(syncp, 0, 4 * sizeof(int) + 256 * sizeof(float), stream);

  dim3 g1(NWG, 2);
  lstm_scan_kernel<<<g1, 256, 0, stream>>>(sent, lens, emb,
                                           Wih_f, Whh_f, b_f,
                                           Wih_b, Whh_b, b_b,
                                           h_f, h_b, h_st,
                                           syncp, syncp + 1, zpad);
  em_kernel<<<(B * T) / 8, 256, 0, stream>>>(h_f, h_b, lens, W_out, b_out, em);
  score_kernel<<<B / 8, 256, 0, stream>>>(tags, lens, em, start, endv, trans, score);
  crf_alpha_kernel<<<B / 32, 288, 0, stream>>>(em, lens, start, endv, trans, logZ);
  final_kernel<<<1, 128, 0, stream>>>(score, logZ, (float*)d_out);
}